// AntiAliasInterpolation2d_5471788335422
// MI455X (gfx1250) — compile-verified
//
#include <hip/hip_runtime.h>

typedef float v2f __attribute__((ext_vector_type(2)));
typedef float v8f __attribute__((ext_vector_type(8)));

#define H   512
#define W   512
#define OH  128
#define OW  128
#define NC  96           // 32 batch * 3 channels
#define TILES_PER_IMG 64 // 8x8 tiles of 16x16 outputs
#define WAVES_PER_BLOCK 4

// Pass 1 (horizontal blur + 4x decimate) for one wave's 80x16 intermediate
// tile, expressed as band-matrix GEMM on V_WMMA_F32_16X16X4_F32.
// EDGE=false: tile fully inside the image -> no clamps/selects, loads use
// immediate offsets off a single per-subtile base address.
template <bool EDGE>
__device__ __forceinline__ void pass1(const float* __restrict__ src,
                                      int ybase, int xbase,
                                      float (*__restrict__ T)[84],
                                      const float (*__restrict__ Hm)[80],
                                      int l15, int half)
{
    #pragma unroll
    for (int sub = 0; sub < 5; ++sub) {
        v8f acc = {};
        const int y = ybase + 16 * sub + l15;      // A-matrix row (M = l15)
        bool yv = true;
        const float* rowp;
        if (EDGE) {
            yv = (y >= 0 && y < H);
            const int yc = min(max(y, 0), H - 1);
            rowp = src + (size_t)yc * W;           // x added per-kk (clamped)
        } else {
            rowp = src + (size_t)y * W + xbase + 2 * half;  // all-in-range base
        }
        #pragma unroll
        for (int kk = 0; kk < 19; ++kk) {
            v2f a;
            if (EDGE) {
                int x  = xbase + 4 * kk + 2 * half;   // even -> 8B aligned
                int xc = min(max(x, 0), W - 2);
                a = *(const v2f*)(rowp + xc);
                a.x = (yv && (unsigned)x       < (unsigned)W) ? a.x : 0.0f;
                a.y = (yv && (unsigned)(x + 1) < (unsigned)W) ? a.y : 0.0f;
            } else {
                a = *(const v2f*)(rowp + 4 * kk);     // immediate IOFFSET = 16*kk
            }
            v2f b = *(const v2f*)(&Hm[l15][4 * kk + 2 * half]);  // band matrix
            acc = __builtin_amdgcn_wmma_f32_16x16x4_f32(false, a, false, b,
                                                        (short)0, acc, false, false);
        }
        // D layout: VGPR r holds (M = r + 8*half, N = l15); store transposed
        #pragma unroll
        for (int r = 0; r < 8; ++r)
            T[l15][16 * sub + r + 8 * half] = acc[r];
    }
}

// Fused separable Gaussian downsample, one 16x16 output tile per wave32.
// The 80x16 intermediate lives only in LDS (transposed, stride 84 to avoid
// bank conflicts). Both passes are band-matrix GEMMs on the same LDS table.
__global__ __launch_bounds__(128)
void aa_interp_wmma_kernel(const float* __restrict__ in,
                           const float* __restrict__ wgt,
                           float* __restrict__ out)
{
    __shared__ float ctab[13];
    __shared__ float Hm[16][80];                  // Band[i][w] = c[w-4i]
    __shared__ float Tt[WAVES_PER_BLOCK][16][84]; // transposed intermediate

    const int tid = threadIdx.x;

    // 1D normalized taps = row sums of the rank-1 normalized 13x13 weight.
    if (tid < 13) {
        float s = 0.f;
        #pragma unroll
        for (int j = 0; j < 13; ++j) s += wgt[tid * 13 + j];
        ctab[tid] = s;
    }
    __syncthreads();
    for (int idx = tid; idx < 16 * 80; idx += 128) {
        int n = idx / 80, w = idx % 80;
        int d = w - 4 * n;
        Hm[n][w] = (d >= 0 && d < 13) ? ctab[d] : 0.0f;
    }
    __syncthreads();

    const int wv   = tid >> 5;
    const int lane = tid & 31;
    const int half = lane >> 4;  // K-split half of the wave
    const int l15  = lane & 15;

    const int tileId = blockIdx.x * WAVES_PER_BLOCK + wv;   // exact: 6144 tiles
    const int nc  = tileId / TILES_PER_IMG;
    const int t   = tileId % TILES_PER_IMG;
    const int oy0 = (t >> 3) * 16;
    const int ox0 = (t & 7) * 16;

    const float* src = in + (size_t)nc * (H * W);
    const int ybase = 4 * oy0 - 6;   // first intermediate row needed
    const int xbase = 4 * ox0 - 6;   // first input column needed

    float (*T)[84] = Tt[wv];

    // ---- pass 1: wave-uniform interior/edge specialization ----
    // interior iff y in [ybase, ybase+79] subset [0,H) and x in
    // [xbase, xbase+75] subset [0,W)  <=>  oy0,ox0 in [16,96]
    const bool interior = (oy0 >= 16) & (oy0 <= 96) & (ox0 >= 16) & (ox0 <= 96);
    if (interior)
        pass1<false>(src, ybase, xbase, T, Hm, l15, half);
    else
        pass1<true>(src, ybase, xbase, T, Hm, l15, half);
    // same-wave LDS ops are in-order: pass-2 reads below need no barrier

    // ---- pass 2: vertical blur + decimate T -> 16x16 output tile ----
    v8f o = {};
    #pragma unroll
    for (int kk = 0; kk < 19; ++kk) {
        v2f a = *(const v2f*)(&Hm[l15][4 * kk + 2 * half]); // Band[m][w]
        v2f b = *(const v2f*)(&T [l15][4 * kk + 2 * half]); // T[w][n] (transposed)
        o = __builtin_amdgcn_wmma_f32_16x16x4_f32(false, a, false, b,
                                                  (short)0, o, false, false);
    }

    float* dst = out + (size_t)nc * (OH * OW);
    #pragma unroll
    for (int r = 0; r < 8; ++r)
        dst[(size_t)(oy0 + r + 8 * half) * OW + (ox0 + l15)] = o[r];
}

extern "C" void kernel_launch(void* const* d_in, const int* in_sizes, int n_in,
                              void* d_out, int out_size, void* d_ws, size_t ws_size,
                              hipStream_t stream) {
    const float* inp = (const float*)d_in[0];   // [32,3,512,512] f32
    const float* wgt = (const float*)d_in[1];   // [3,1,13,13] f32
    float* out = (float*)d_out;                 // [32,3,128,128] f32

    dim3 grid(NC * TILES_PER_IMG / WAVES_PER_BLOCK);  // 1536 blocks
    dim3 block(128);                                  // 4 wave32 per block
    aa_interp_wmma_kernel<<<grid, block, 0, stream>>>(inp, wgt, out);
}